// OpticalLayer_7550552507221
// MI455X (gfx1250) — compile-verified
//
#include <hip/hip_runtime.h>

typedef __attribute__((ext_vector_type(2))) float v2f;
typedef __attribute__((ext_vector_type(8))) float v8f;

namespace {
constexpr int kWH = 65536;            // W*H = 256*256
constexpr int kK  = 512;              // number of filters
constexpr int kB  = 8;                // batch
constexpr int kKT = kK / 16;          // 32 k-tiles of 16 rows
constexpr int kChunk = 1024;          // i-elements per phase-1 wave
constexpr int kNCH = kWH / kChunk;    // 64 chunks
// workspace layout (float offsets)
constexpr size_t P_OFF   = 0;                                   // partials: 32*64*256 = 524288
constexpr size_t Y_OFF   = (size_t)kKT * kNCH * 256;            // y[k][16]: 8192
constexpr size_t XR_OFF  = Y_OFF + (size_t)kK * 16;             // xr[8][65536]: 524288
constexpr size_t MAX_OFF = XR_OFF + (size_t)kB * kWH;           // 1 uint max cell
}

// jnp.sign semantics: -1, 0, +1 (exact zero maps to 0)
__device__ __forceinline__ float sgn(float w) {
  float pm1 = __uint_as_float((__float_as_uint(w) & 0x80000000u) | 0x3f800000u);
  return (w == 0.0f) ? 0.0f : pm1;
}

// ---------------- Phase 1: y partials = sign(W)(16k x i-chunk) * X^T ----------------
// D(16x16) = A(16x4: k-rows x i) * B(4x16: i x b), accumulated over the wave's i-chunk.
// Columns b>=8 of D receive duplicated-x garbage; they are never consumed downstream
// (phase-2 rows 8..15 are discarded), so no masking is needed in the hot loop.
__global__ __launch_bounds__(256)
void phase1(const float* __restrict__ x, const float* __restrict__ w,
            float* __restrict__ ws) {
  const int wave = blockIdx.x * 8 + (threadIdx.x >> 5);
  const int lane = threadIdx.x & 31;
  const int kt = wave / kNCH;            // 0..31
  const int ch = wave - kt * kNCH;       // 0..63
  const int lm = lane & 15;
  const int o  = (lane < 16) ? 0 : 2;    // K-pair offset for this lane half

  // A row: one k-row of weights per lane (documented 16x4 A layout)
  const float* __restrict__ arow = w + (size_t)(kt * 16 + lm) * kWH;
  // B: x row for n = lm (dup of lm&7 for n>=8; dead columns)
  const float* __restrict__ brow = x + (size_t)(lm & 7) * kWH;

  v8f acc0 = {};
  v8f acc1 = {};
  const int ibeg = ch * kChunk;
  #pragma unroll 2
  for (int i0 = ibeg; i0 < ibeg + kChunk; i0 += 8) {
    // chain 0
    v2f aw0 = *(const v2f*)(arow + i0 + o);          // global_load_b64
    v2f bx0 = *(const v2f*)(brow + i0 + o);          // L2-hot (x is 2 MB)
    v2f A0; A0.x = sgn(aw0.x); A0.y = sgn(aw0.y);
    acc0 = __builtin_amdgcn_wmma_f32_16x16x4_f32(
        false, A0, false, bx0, (short)0, acc0, false, false);
    // chain 1 (independent accumulator to avoid back-to-back WMMA RAW)
    v2f aw1 = *(const v2f*)(arow + i0 + 4 + o);
    v2f bx1 = *(const v2f*)(brow + i0 + 4 + o);
    v2f A1; A1.x = sgn(aw1.x); A1.y = sgn(aw1.y);
    acc1 = __builtin_amdgcn_wmma_f32_16x16x4_f32(
        false, A1, false, bx1, (short)0, acc1, false, false);
  }

  // Store full 16x16 partial tile deterministically (no float atomics).
  float* __restrict__ p = ws + P_OFF + (size_t)wave * 256;
  const int rbase = (lane < 16) ? 0 : 8;             // C/D layout: M = r or r+8
  #pragma unroll
  for (int r = 0; r < 8; ++r)
    p[(rbase + r) * 16 + lm] = acc0[r] + acc1[r];
}

// ---------------- Reduce partials -> y[k][16]; reset max cell ----------------
__global__ __launch_bounds__(256)
void reduce_y(float* __restrict__ ws) {
  const int t = blockIdx.x * 256 + threadIdx.x;    // t = k*16 + b, 0..8191
  const int kt  = t >> 8;                          // k / 16
  const int idx = t & 255;                         // (k%16)*16 + b
  const float* __restrict__ p = ws + P_OFF + (size_t)kt * kNCH * 256 + idx;
  float s = 0.0f;
  #pragma unroll 4
  for (int ch = 0; ch < kNCH; ++ch) s += p[(size_t)ch * 256];
  ws[Y_OFF + t] = s;
  if (t == 0) ((unsigned*)ws)[MAX_OFF] = 0u;       // identity for mapped-float max
}

// ---------------- Phase 2: xr(16b x 16i tile) = Y(16x512) * sign(W)(512 x 16i) ------
__global__ __launch_bounds__(256)
void phase2(const float* __restrict__ w, float* __restrict__ ws) {
  __shared__ float y[kK * 16];                     // 32 KB, shared by 8 waves
  for (int t = threadIdx.x; t < kK * 16; t += 256) y[t] = ws[Y_OFF + t];
  __syncthreads();

  const int lane = threadIdx.x & 31;
  const int lm = lane & 15;
  const int o  = (lane < 16) ? 0 : 2;
  const int tile = blockIdx.x * 8 + (threadIdx.x >> 5);   // 0..4095
  const int i0 = tile * 16;

  v8f acc0 = {};
  v8f acc1 = {};
  #pragma unroll 2
  for (int k0 = 0; k0 < kK; k0 += 8) {
    // chain 0
    {
      v2f A = { y[(k0 + o) * 16 + lm], y[(k0 + o + 1) * 16 + lm] };
      const float* bp = w + (size_t)(k0 + o) * kWH + (i0 + lm);
      v2f Bv = { sgn(bp[0]), sgn(bp[kWH]) };      // rows k0+o, k0+o+1 (64B segments)
      acc0 = __builtin_amdgcn_wmma_f32_16x16x4_f32(
          false, A, false, Bv, (short)0, acc0, false, false);
    }
    // chain 1
    {
      v2f A = { y[(k0 + 4 + o) * 16 + lm], y[(k0 + 4 + o + 1) * 16 + lm] };
      const float* bp = w + (size_t)(k0 + 4 + o) * kWH + (i0 + lm);
      v2f Bv = { sgn(bp[0]), sgn(bp[kWH]) };
      acc1 = __builtin_amdgcn_wmma_f32_16x16x4_f32(
          false, A, false, Bv, (short)0, acc1, false, false);
    }
  }

  // Rows 0..7 (= real b) live in lanes 0..15, VGPRs 0..7.
  float m = -__builtin_inff();
  if (lane < 16) {
    #pragma unroll
    for (int r = 0; r < 8; ++r) {
      const float v = acc0[r] + acc1[r];
      ws[XR_OFF + (size_t)r * kWH + i0 + lm] = v;
      m = fmaxf(m, v);
    }
  }
  // wave32 max reduction, then exact (order-independent) uint atomicMax
  #pragma unroll
  for (int off = 16; off; off >>= 1)
    m = fmaxf(m, __shfl_xor(m, off, 32));
  if (lane == 0) {
    unsigned u = __float_as_uint(m);
    unsigned key = (u & 0x80000000u) ? ~u : (u | 0x80000000u);
    atomicMax(((unsigned*)ws) + MAX_OFF, key);
  }
}

// ---------------- Phase 3: out = xr / max(xr) ----------------
__global__ __launch_bounds__(256)
void phase3(const float* __restrict__ ws, float* __restrict__ out) {
  const size_t t = (size_t)blockIdx.x * 256 + threadIdx.x;
  const unsigned key  = ((const unsigned*)ws)[MAX_OFF];
  const unsigned bits = (key & 0x80000000u) ? (key & 0x7fffffffu) : ~key;
  const float mx = __uint_as_float(bits);
  out[t] = ws[XR_OFF + t] / mx;
}

extern "C" void kernel_launch(void* const* d_in, const int* in_sizes, int n_in,
                              void* d_out, int out_size, void* d_ws, size_t ws_size,
                              hipStream_t stream) {
  const float* x = (const float*)d_in[0];   // (8,1,256,256) f32
  const float* w = (const float*)d_in[1];   // (512,256,256) f32
  float* out = (float*)d_out;               // (8,1,256,256) f32
  float* ws  = (float*)d_ws;

  const int p1_waves = kKT * kNCH;                       // 2048 waves
  phase1 <<<p1_waves / 8,        256, 0, stream>>>(x, w, ws);
  reduce_y<<<(kK * 16) / 256,    256, 0, stream>>>(ws);  // 32 blocks
  phase2 <<<(kWH / 16) / 8,      256, 0, stream>>>(w, ws); // 512 blocks
  phase3 <<<(kB * kWH) / 256,    256, 0, stream>>>(ws, out); // 2048 blocks
}